// GCNEncoder_42666205118859
// MI455X (gfx1250) — compile-verified
//
#include <hip/hip_runtime.h>

typedef float v2f __attribute__((ext_vector_type(2)));
typedef float v8f __attribute__((ext_vector_type(8)));

#define NNODES  131072
#define NEDGES  2097152
#define NGRAPH  64
#define DIN     64
#define HDIM    128
#define GPDIM   32
#define GDIMV   6
#define BNEPS   1e-5f
#define MT      64        // rows per block in the WMMA GEMMs (4 row tiles)

// ---------------------------------------------------------------- zero fill
__global__ void k_zero(float* __restrict__ p, long n) {
  long i = (long)blockIdx.x * blockDim.x + threadIdx.x;
  if (i < n) p[i] = 0.0f;
}

// ---------------------------------------------------------------- degree
__global__ void k_degree(const int* __restrict__ rowi, float* __restrict__ deg, int E) {
  int e = blockIdx.x * blockDim.x + threadIdx.x;
  if (e < E) atomicAdd(&deg[rowi[e]], 1.0f);
}

// ---------------------------------------------------------------- h = x @ W_in + b_in  (WMMA f32 16x16x4)
// block: 256 thr = 8 waves; block owns a 64-row stripe (4 row tiles); wave w owns column tile w.
// A tile is a raw contiguous copy of x rows -> staged with GLOBAL_LOAD_ASYNC_TO_LDS_B128 (ASYNCcnt).
__global__ void k_in_gemm(const float* __restrict__ x, const float* __restrict__ W,
                          const float* __restrict__ bias, float* __restrict__ hout) {
  __shared__ float At[MT * DIN];                    // 16 KB A tile (64x64)
  const int t = threadIdx.x;
  const int row0 = blockIdx.x * MT;

  // async stage: 64*64 floats = 1024 x b128 chunks; 256 threads x 4 (tile is fully contiguous in x)
#pragma unroll
  for (int i = 0; i < 4; ++i) {
    int c4 = i * 256 + t;                           // float4 index within tile
    unsigned loff = (unsigned)(unsigned long long)(&At[c4 * 4]);   // low 32 bits of LDS aperture = LDS offset
    const float* gsrc = x + (size_t)row0 * DIN + (size_t)c4 * 4;
    asm volatile("global_load_async_to_lds_b128 %0, %1, off" :: "v"(loff), "v"(gsrc) : "memory");
  }
  asm volatile("s_wait_asynccnt 0x0" ::: "memory");
  __syncthreads();

  const int wave = t >> 5, lane = t & 31;
  const int gq = lane >> 4, ln = lane & 15;
  const int n = wave * 16 + ln;                     // output column
  v8f acc[4];
#pragma unroll
  for (int rt = 0; rt < 4; ++rt) acc[rt] = (v8f){0, 0, 0, 0, 0, 0, 0, 0};
#pragma unroll
  for (int k = 0; k < DIN; k += 4) {
    int ka = k + 2 * gq;                            // K pair for this lane half
    v2f b;
    b.x = W[(size_t)ka * HDIM + n];                 // B: lane = N, VGPRs = K pair; reused by 4 row tiles
    b.y = W[(size_t)(ka + 1) * HDIM + n];
#pragma unroll
    for (int rt = 0; rt < 4; ++rt) {                // 4 independent accumulator chains
      v2f a = *(const v2f*)(At + (rt * 16 + ln) * DIN + ka);
      acc[rt] = __builtin_amdgcn_wmma_f32_16x16x4_f32(false, a, false, b, (short)0, acc[rt], false, false);
    }
  }
  float bz = bias[n];
#pragma unroll
  for (int rt = 0; rt < 4; ++rt)
#pragma unroll
    for (int r = 0; r < 8; ++r)                     // C/D: VGPR r -> M = r + 8*gq, N = ln
      hout[(size_t)(row0 + rt * 16 + r + 8 * gq) * HDIM + n] = acc[rt][r] + bz;
}

// ---------------------------------------------------------------- edge message + scatter-add
// agg[row[e]][:] += h[col[e]][:] * (er0*We[0][:] + er1*We[1][:] + be[:])
// one wave per 8 edges; lane covers 4 channels; ea recomputed on the fly (never materialized);
// h rows prefetched up-front to overlap the random L2 gathers.
__global__ void k_scatter(const float* __restrict__ h, const float* __restrict__ er,
                          const int* __restrict__ rowi, const int* __restrict__ coli,
                          const float* __restrict__ We, const float* __restrict__ be,
                          float* __restrict__ agg) {
  const int lane = threadIdx.x & 31;
  const int wid = blockIdx.x * (blockDim.x >> 5) + (threadIdx.x >> 5);
  const int j = lane << 2;                          // channel base: 4 floats per lane = 128
  const float4 w0 = *(const float4*)(We + j);
  const float4 w1 = *(const float4*)(We + HDIM + j);
  const float4 bb = *(const float4*)(be + j);
  const int e0 = wid * 8;

  int cs[8], rs[8];
  float a0s[8], a1s[8];
#pragma unroll
  for (int q = 0; q < 8; ++q) {
    int e = e0 + q;
    cs[q] = coli[e];
    rs[q] = rowi[e];
    a0s[q] = er[2 * e + 0];
    a1s[q] = er[2 * e + 1];
  }
#pragma unroll
  for (int q = 0; q < 8; ++q)                       // global_prefetch_b8 the gathered rows
    __builtin_prefetch(h + (size_t)cs[q] * HDIM + j, 0, 0);

#pragma unroll
  for (int q = 0; q < 8; ++q) {
    float4 hv = *(const float4*)(h + (size_t)cs[q] * HDIM + j);
    float4 w;
    w.x = fmaf(a0s[q], w0.x, fmaf(a1s[q], w1.x, bb.x));
    w.y = fmaf(a0s[q], w0.y, fmaf(a1s[q], w1.y, bb.y));
    w.z = fmaf(a0s[q], w0.z, fmaf(a1s[q], w1.z, bb.z));
    w.w = fmaf(a0s[q], w0.w, fmaf(a1s[q], w1.w, bb.w));
    float* ap = agg + (size_t)rs[q] * HDIM + j;
    atomicAdd(ap + 0, hv.x * w.x);
    atomicAdd(ap + 1, hv.y * w.y);
    atomicAdd(ap + 2, hv.z * w.z);
    atomicAdd(ap + 3, hv.w * w.w);
  }
}

// ---------------------------------------------------------------- hout = relu(bn((h + agg/deg) @ W + b))
// 64-row stripe; fused A tile staged through LDS with float4 loads; B reused across 4 row tiles.
__global__ void k_layer_gemm(const float* __restrict__ hin, const float* __restrict__ agg,
                             const float* __restrict__ deg, const float* __restrict__ W,
                             const float* __restrict__ bias, const float* __restrict__ gamma,
                             const float* __restrict__ beta, const float* __restrict__ mean,
                             const float* __restrict__ var, float* __restrict__ hout) {
  __shared__ float At[MT * HDIM];                   // 32 KB fused A tile (64x128)
  __shared__ float invd[MT];
  const int t = threadIdx.x;
  const int row0 = blockIdx.x * MT;
  if (t < MT) invd[t] = 1.0f / fmaxf(deg[row0 + t], 1.0f);
  __syncthreads();
  // stage fused residual+agg tile: 8192 floats = 2048 float4; 256 threads x 8
#pragma unroll
  for (int i = 0; i < 8; ++i) {
    int c4 = i * 256 + t;                           // float4 index
    int r = c4 >> 5;                                // 32 float4 per row
    size_t off = (size_t)(row0 + r) * HDIM + (size_t)(c4 & 31) * 4;
    float4 hv = *(const float4*)(hin + off);
    float4 av = *(const float4*)(agg + off);
    float iv = invd[r];
    float4 o;
    o.x = fmaf(av.x, iv, hv.x);
    o.y = fmaf(av.y, iv, hv.y);
    o.z = fmaf(av.z, iv, hv.z);
    o.w = fmaf(av.w, iv, hv.w);
    *(float4*)(At + c4 * 4) = o;
  }
  __syncthreads();

  const int wave = t >> 5, lane = t & 31;
  const int gq = lane >> 4, ln = lane & 15;
  const int n = wave * 16 + ln;
  v8f acc[4];
#pragma unroll
  for (int rt = 0; rt < 4; ++rt) acc[rt] = (v8f){0, 0, 0, 0, 0, 0, 0, 0};
#pragma unroll
  for (int k = 0; k < HDIM; k += 4) {
    int ka = k + 2 * gq;
    v2f b;
    b.x = W[(size_t)ka * HDIM + n];
    b.y = W[(size_t)(ka + 1) * HDIM + n];
#pragma unroll
    for (int rt = 0; rt < 4; ++rt) {
      v2f a = *(const v2f*)(At + (rt * 16 + ln) * HDIM + ka);
      acc[rt] = __builtin_amdgcn_wmma_f32_16x16x4_f32(false, a, false, b, (short)0, acc[rt], false, false);
    }
  }
  // fused bias + eval BatchNorm + ReLU (per-column scale/shift)
  float sc = gamma[n] * rsqrtf(var[n] + BNEPS);
  float sh = beta[n] - mean[n] * sc;
  float bz = bias[n];
#pragma unroll
  for (int rt = 0; rt < 4; ++rt)
#pragma unroll
    for (int r = 0; r < 8; ++r) {
      float z = (acc[rt][r] + bz) * sc + sh;
      hout[(size_t)(row0 + rt * 16 + r + 8 * gq) * HDIM + n] = fmaxf(z, 0.0f);
    }
}

// ---------------------------------------------------------------- mean-pool: exploits sorted `batch`
__global__ void k_pool(const float* __restrict__ h, const int* __restrict__ batch,
                       float* __restrict__ gsum, float* __restrict__ gcnt, int N) {
  const int NPB = 256;
  const int n0 = blockIdx.x * NPB;
  const int j = threadIdx.x;
  float acc = 0.0f;
  int cnt = 0;
  int cur = batch[n0];
  for (int i = 0; i < NPB; ++i) {
    int n = n0 + i;
    if (n >= N) break;
    int g = batch[n];
    if (g != cur) {
      atomicAdd(&gsum[cur * HDIM + j], acc);
      if (j == 0) atomicAdd(&gcnt[cur], (float)cnt);
      acc = 0.0f; cnt = 0; cur = g;
    }
    acc += h[(size_t)n * HDIM + j];
    ++cnt;
  }
  atomicAdd(&gsum[cur * HDIM + j], acc);
  if (j == 0) atomicAdd(&gcnt[cur], (float)cnt);
}

// ---------------------------------------------------------------- gp = relu(global_x @ W_gproj + b)  (tiny)
__global__ void k_gproj(const float* __restrict__ gx, const float* __restrict__ W,
                        const float* __restrict__ b, float* __restrict__ gp) {
  int idx = blockIdx.x * blockDim.x + threadIdx.x;
  if (idx < NGRAPH * GPDIM) {
    int gi = idx >> 5, j = idx & 31;
    float acc = b[j];
#pragma unroll
    for (int k = 0; k < GDIMV; ++k) acc = fmaf(gx[gi * GDIMV + k], W[k * GPDIM + j], acc);
    gp[idx] = fmaxf(acc, 0.0f);
  }
}

// ---------------------------------------------------------------- out = concat(gsum/gcnt, gp) @ W_gcomb + b
__global__ void k_final(const float* __restrict__ gsum, const float* __restrict__ gcnt,
                        const float* __restrict__ gp, const float* __restrict__ W,
                        const float* __restrict__ bias, float* __restrict__ out) {
  const int t = threadIdx.x;
  const int wave = t >> 5, lane = t & 31;
  const int gq = lane >> 4, ln = lane & 15;
  const int n = wave * 16 + ln;
  const float bz = bias[n];
#pragma unroll
  for (int rt = 0; rt < 4; ++rt) {                  // 64 graphs = 4 row tiles
    const int m = rt * 16 + ln;
    const float invc = 1.0f / fmaxf(gcnt[m], 1.0f);
    v8f acc = {0, 0, 0, 0, 0, 0, 0, 0};
#pragma unroll
    for (int k = 0; k < HDIM + GPDIM; k += 4) {
      int ka = k + 2 * gq;                          // pair never straddles the 128 boundary (k % 4 == 0)
      v2f a, b;
      a.x = (ka     < HDIM) ? gsum[m * HDIM + ka]     * invc : gp[m * GPDIM + (ka - HDIM)];
      a.y = (ka + 1 < HDIM) ? gsum[m * HDIM + ka + 1] * invc : gp[m * GPDIM + (ka + 1 - HDIM)];
      b.x = W[ka * HDIM + n];
      b.y = W[(ka + 1) * HDIM + n];
      acc = __builtin_amdgcn_wmma_f32_16x16x4_f32(false, a, false, b, (short)0, acc, false, false);
    }
#pragma unroll
    for (int r = 0; r < 8; ++r) {
      int m2 = rt * 16 + r + 8 * gq;
      out[m2 * HDIM + n] = acc[r] + bz;
    }
  }
}

// ================================================================ host
extern "C" void kernel_launch(void* const* d_in, const int* in_sizes, int n_in,
                              void* d_out, int out_size, void* d_ws, size_t ws_size,
                              hipStream_t stream) {
  (void)in_sizes; (void)n_in; (void)out_size; (void)ws_size;
  const float* x      = (const float*)d_in[0];
  const float* er     = (const float*)d_in[1];
  const float* gx     = (const float*)d_in[2];
  const float* W_in   = (const float*)d_in[3];
  const float* b_in   = (const float*)d_in[4];
  const float* W_edge = (const float*)d_in[5];
  const float* b_edge = (const float*)d_in[6];
  const float* W_gcn  = (const float*)d_in[7];
  const float* b_gcn  = (const float*)d_in[8];
  const float* bn_g   = (const float*)d_in[9];
  const float* bn_b   = (const float*)d_in[10];
  const float* bn_m   = (const float*)d_in[11];
  const float* bn_v   = (const float*)d_in[12];
  const float* W_gp   = (const float*)d_in[13];
  const float* b_gp   = (const float*)d_in[14];
  const float* W_gc   = (const float*)d_in[15];
  const float* b_gc   = (const float*)d_in[16];
  const int*   rowi   = (const int*)d_in[17];
  const int*   coli   = (const int*)d_in[18];
  const int*   batch  = (const int*)d_in[19];
  float* out = (float*)d_out;

  // workspace layout (floats): h_a | h_b | agg | deg | gsum | gcnt | gp  (~193 MB)
  float* ws   = (float*)d_ws;
  const long NH = (long)NNODES * HDIM;
  float* h_a  = ws;
  float* h_b  = h_a + NH;
  float* agg  = h_b + NH;
  float* deg  = agg + NH;               // N
  float* gsum = deg + NNODES;           // G*H
  float* gcnt = gsum + NGRAPH * HDIM;   // G
  float* gp   = gcnt + NGRAPH;          // G*GP

  // zero deg + gsum + gcnt (contiguous)
  {
    long n = NNODES + NGRAPH * HDIM + NGRAPH;
    k_zero<<<(int)((n + 255) / 256), 256, 0, stream>>>(deg, n);
  }
  k_degree<<<NEDGES / 256, 256, 0, stream>>>(rowi, deg, NEDGES);
  k_in_gemm<<<NNODES / MT, 256, 0, stream>>>(x, W_in, b_in, h_a);

  float* hcur = h_a;
  float* hnxt = h_b;
  for (int l = 0; l < 3; ++l) {
    k_zero<<<(int)(NH / 256), 256, 0, stream>>>(agg, NH);
    k_scatter<<<NEDGES / 64, 256, 0, stream>>>(hcur, er, rowi, coli, W_edge, b_edge, agg);
    k_layer_gemm<<<NNODES / MT, 256, 0, stream>>>(
        hcur, agg, deg, W_gcn + (size_t)l * HDIM * HDIM, b_gcn + l * HDIM,
        bn_g + l * HDIM, bn_b + l * HDIM, bn_m + l * HDIM, bn_v + l * HDIM, hnxt);
    float* tmp = hcur; hcur = hnxt; hnxt = tmp;
  }

  k_pool<<<NNODES / 256, 128, 0, stream>>>(hcur, batch, gsum, gcnt, NNODES);
  k_gproj<<<(NGRAPH * GPDIM + 255) / 256, 256, 0, stream>>>(gx, W_gp, b_gp, gp);
  k_final<<<1, 256, 0, stream>>>(gsum, gcnt, gp, W_gc, b_gc, out);
}